// ChamferLoss_40673340293144
// MI455X (gfx1250) — compile-verified
//
#include <hip/hip_runtime.h>
#include <math.h>

typedef __attribute__((ext_vector_type(2))) float v2f;
typedef __attribute__((ext_vector_type(8))) float v8f;

#define N_PTS 8192
#define COLS_PER_BLOCK 1024
#define STRIPS 4            // 4 x 16 rows per wave = 64 rows
#define WAVES_PER_BLOCK 8

// ---------------------------------------------------------------------------
// Kernel 1: pack points as (x, y, z, ||p||^2) and init min arrays to +inf
// ---------------------------------------------------------------------------
__global__ void pack_init_kernel(const float* __restrict__ tgt,
                                 const float* __restrict__ outp,
                                 float4* __restrict__ Pt, float4* __restrict__ Po,
                                 unsigned* __restrict__ minT, unsigned* __restrict__ minO) {
    int i = blockIdx.x * blockDim.x + threadIdx.x;
    if (i < N_PTS) {
        float x = tgt[3 * i], y = tgt[3 * i + 1], z = tgt[3 * i + 2];
        Pt[i] = make_float4(x, y, z, x * x + y * y + z * z);
        x = outp[3 * i]; y = outp[3 * i + 1]; z = outp[3 * i + 2];
        Po[i] = make_float4(x, y, z, x * x + y * y + z * z);
        minT[i] = 0x7F800000u;   // +inf as uint (nonneg float order == uint order)
        minO[i] = 0x7F800000u;
    }
}

// ---------------------------------------------------------------------------
// Kernel 2: WMMA distance tiles + running row-mins.
//   D = A x B + C with A row m = (ax, ay, az, 1),
//                      B col n = (-2bx, -2by, -2bz, ||b||^2),
//                      C[m][*] = ||a_m||^2          (loop-invariant!)
//   => D[m][n] = ||a||^2 + ||b||^2 - 2 a.b
// grid.x = 256: dir(2) x rowBlock(16) x colChunk(8)
// ---------------------------------------------------------------------------
__global__ __launch_bounds__(256) void chamfer_tiles_kernel(
        const float4* __restrict__ Pt, const float4* __restrict__ Po,
        unsigned* __restrict__ minT, unsigned* __restrict__ minO) {
    __shared__ float sB[COLS_PER_BLOCK * 4];   // 16 KB: staged column points

    const int b        = blockIdx.x;
    const int dir      = b >> 7;          // 0: rows=target, 1: rows=output
    const int rem      = b & 127;
    const int rowBlock = rem >> 3;        // 0..15
    const int colChunk = rem & 7;         // 0..7

    const float4* Prow  = dir ? Po : Pt;
    const float4* Pcol  = dir ? Pt : Po;
    unsigned*     minAr = dir ? minO : minT;

    // Cooperative stage of this block's 1024 columns into LDS
    {
        const float4* src = Pcol + colChunk * COLS_PER_BLOCK;
        float4*       dst = (float4*)sB;
        for (int k = threadIdx.x; k < COLS_PER_BLOCK; k += blockDim.x)
            dst[k] = src[k];
    }
    __syncthreads();

    const int wave = threadIdx.x >> 5;    // 0..7
    const int lane = threadIdx.x & 31;
    const int half = lane >> 4;           // 0: K={0,1}, 1: K={2,3}
    const int l16  = lane & 15;
    const int klo  = half ? 2 : 0;
    const float s1m = half ? 1.0f : -2.0f;  // B second-component multiplier

    const int rowBase = rowBlock * (WAVES_PER_BLOCK * STRIPS * 16) + wave * (STRIPS * 16);

    // A operands (row m = (x,y,z,1)) and loop-invariant C (= ||a_m||^2 per row)
    v2f a[STRIPS];
    v8f cm[STRIPS];
#pragma unroll
    for (int s = 0; s < STRIPS; ++s) {
        const float* pm = (const float*)(Prow + (rowBase + s * 16 + l16));
        const float r0  = pm[klo];
        const float r1  = pm[klo + 1];       // half=1 lanes: this is ||a||^2
        a[s][0] = r0;
        a[s][1] = half ? 1.0f : r1;
        // C[vgpr k] = ||a||^2 of row (rowBase + s*16 + k + half*8); lane 16+r
        // holds ||a||^2 of row rowBase+s*16+r in r1 -> fetch via bpermute.
#pragma unroll
        for (int k = 0; k < 8; ++k)
            cm[s][k] = __shfl(r1, 16 + k + half * 8, 32);
    }

    v8f rowmin[STRIPS];
#pragma unroll
    for (int s = 0; s < STRIPS; ++s)
#pragma unroll
        for (int k = 0; k < 8; ++k) rowmin[s][k] = 3.4e38f;

    // Sweep 64 column tiles (2 per iteration); each B tile feeds 4 WMMAs
    for (int t = 0; t < COLS_PER_BLOCK / 16; t += 2) {
        const float* p0 = &sB[(t * 16 + l16) * 4];
        const float* p1 = p0 + 64;                  // next 16-column tile
        v2f b0, b1;
        b0[0] = -2.0f * p0[klo];
        b0[1] = s1m   * p0[klo + 1];
        b1[0] = -2.0f * p1[klo];
        b1[1] = s1m   * p1[klo + 1];

#pragma unroll
        for (int s = 0; s < STRIPS; ++s) {
            v8f d0 = __builtin_amdgcn_wmma_f32_16x16x4_f32(
                false, a[s], false, b0, (short)0, cm[s], false, false);
            v8f d1 = __builtin_amdgcn_wmma_f32_16x16x4_f32(
                false, a[s], false, b1, (short)0, cm[s], false, false);
            // rowmin = min3(rowmin, d0, d1); clamp-to-0 deferred to the end
            // (max(.,0) is monotone, commutes with min). Inline asm avoids
            // clang's sNaN canonicalization (v_max v,v) on every element.
#pragma unroll
            for (int k = 0; k < 8; ++k) {
                float r = rowmin[s][k];
                asm("v_min3_num_f32 %0, %1, %2, %3"
                    : "=v"(r)
                    : "v"(r), "v"(d0[k]), "v"(d1[k]));
                rowmin[s][k] = r;
            }
        }
    }

    // Collapse the N dimension: min across each 16-lane half (xor butterfly)
#pragma unroll
    for (int s = 0; s < STRIPS; ++s) {
#pragma unroll
        for (int k = 0; k < 8; ++k) {
            float v = rowmin[s][k];
            v = fminf(v, __shfl_xor(v, 1, 32));
            v = fminf(v, __shfl_xor(v, 2, 32));
            v = fminf(v, __shfl_xor(v, 4, 32));
            v = fminf(v, __shfl_xor(v, 8, 32));
            rowmin[s][k] = fmaxf(v, 0.0f);   // deferred clamp of numerical negatives
        }
    }

    // lanes 0 / 16 publish rows [rowBase+s*16 .. +7] / [+8 .. +15]
    if (l16 == 0) {
#pragma unroll
        for (int s = 0; s < STRIPS; ++s) {
            const int row0 = rowBase + s * 16 + half * 8;
#pragma unroll
            for (int k = 0; k < 8; ++k)
                atomicMin(&minAr[row0 + k], __float_as_uint(rowmin[s][k]));
        }
    }
}

// ---------------------------------------------------------------------------
// Kernel 3: deterministic sqrt-sum reduction + scale
// ---------------------------------------------------------------------------
__global__ __launch_bounds__(1024) void finalize_kernel(
        const unsigned* __restrict__ minT, const unsigned* __restrict__ minO,
        const int* __restrict__ curp, const int* __restrict__ subp,
        float* __restrict__ out) {
    __shared__ float red[32];
    const int tid = threadIdx.x;
    float s = 0.0f;
    for (int i = tid; i < N_PTS; i += 1024) {
        s += sqrtf(__uint_as_float(minT[i]));
        s += sqrtf(__uint_as_float(minO[i]));
    }
#pragma unroll
    for (int m = 1; m < 32; m <<= 1) s += __shfl_xor(s, m, 32);
    if ((tid & 31) == 0) red[tid >> 5] = s;
    __syncthreads();
    if (tid < 32) {
        float v = red[tid];
#pragma unroll
        for (int m = 1; m < 32; m <<= 1) v += __shfl_xor(v, m, 32);
        if (tid == 0) {
            const int cur = curp[0], sub = subp[0];
            const float scale = 10.0f / powf(0.99f, (float)(cur / sub));
            // loss = (mean(sqrt d1) + mean(sqrt d2)) * 0.5 * scale
            out[0] = v * (0.5f / (float)N_PTS) * scale;
        }
    }
}

// ---------------------------------------------------------------------------
extern "C" void kernel_launch(void* const* d_in, const int* in_sizes, int n_in,
                              void* d_out, int out_size, void* d_ws, size_t ws_size,
                              hipStream_t stream) {
    (void)in_sizes; (void)n_in; (void)out_size; (void)ws_size;
    const float* tgt  = (const float*)d_in[0];
    const float* outp = (const float*)d_in[1];
    const int*   cur  = (const int*)d_in[2];
    const int*   sub  = (const int*)d_in[3];

    char* ws = (char*)d_ws;
    float4*   Pt   = (float4*)ws;                                  // 128 KB
    float4*   Po   = (float4*)(ws + (size_t)N_PTS * 16);           // 128 KB
    unsigned* minT = (unsigned*)(ws + 2 * (size_t)N_PTS * 16);     //  32 KB
    unsigned* minO = (unsigned*)(ws + 2 * (size_t)N_PTS * 16 + (size_t)N_PTS * 4);

    pack_init_kernel<<<N_PTS / 256, 256, 0, stream>>>(tgt, outp, Pt, Po, minT, minO);
    chamfer_tiles_kernel<<<256, 256, 0, stream>>>(Pt, Po, minT, minO);
    finalize_kernel<<<1, 1024, 0, stream>>>(minT, minO, cur, sub, (float*)d_out);
}